// SG_RSNN_55448027791474
// MI455X (gfx1250) — compile-verified
//
#include <hip/hip_runtime.h>
#include <math.h>

typedef float v2f __attribute__((ext_vector_type(2)));
typedef float v4f __attribute__((ext_vector_type(4)));
typedef float v8f __attribute__((ext_vector_type(8)));

#define BATCH     8192
#define IN_DIMS   256
#define NN        512
#define OUT_DIMS  128
#define NSTEPS    33

// d_ws layout (float offsets)
#define KH_OFF    0         // 32nt*64cp*32lane v4f = 262144 floats (1 MB)
#define KIN_OFF   262144    // 32nt*32cp*32lane v4f = 131072 floats (512 KB)
#define KOUT_OFF  393216    // 8nt*64cp*32lane  v4f =  65536 floats (256 KB)

static __device__ __forceinline__ v8f v8zero() {
  v8f z = {0.f, 0.f, 0.f, 0.f, 0.f, 0.f, 0.f, 0.f};
  return z;
}

static __device__ __forceinline__ v8f wmma_f32(v2f a, v2f b, v8f c) {
  // D = A(16x4 f32) x B(4x16 f32) + C(16x16 f32); VOP3P V_WMMA_F32_16X16X4_F32
  return __builtin_amdgcn_wmma_f32_16x16x4_f32(
      /*neg_a=*/false, a, /*neg_b=*/false, b,
      /*c_mod=*/(short)0, c, /*reuse_a=*/false, /*reuse_b=*/false);
}

// ---- weight pre-pack: B-fragment order, TWO K-chunks per lane (b128) ----
// B(4x16) frag layout: VGPR0 lanes0-15 -> K=+0, lanes16-31 -> K=+2; VGPR1 -> +1/+3.
// v4f packed[(nt*PAIRS + cp)*32 + lane] =
//   { w[k0][n], w[k0+1][n], w[k0+4][n], w[k0+5][n] },
//   k0 = 8*cp + 2*(lane>>4), n = nt*16 + (lane&15).  Scale folded in.

__global__ __launch_bounds__(256) void pack_kh(const float* __restrict__ kh,
                                               v4f* __restrict__ dst, float R) {
  int i = blockIdx.x * 256 + threadIdx.x;          // 32*64*32 = 65536
  int lane = i & 31, cp = (i >> 5) & 63, nt = i >> 11;
  int k0 = 8 * cp + 2 * (lane >> 4);
  int n = nt * 16 + (lane & 15);
  v4f v;
  v.x = R * kh[(size_t)(k0 + 0) * NN + n];
  v.y = R * kh[(size_t)(k0 + 1) * NN + n];
  v.z = R * kh[(size_t)(k0 + 4) * NN + n];
  v.w = R * kh[(size_t)(k0 + 5) * NN + n];
  dst[i] = v;
}

__global__ __launch_bounds__(256) void pack_kin(const float* __restrict__ kin,
                                                v4f* __restrict__ dst, float Rin) {
  // cat(x,-x)@W == x@(W[:256]-W[256:]) : fold difference + Rin into the pack
  int i = blockIdx.x * 256 + threadIdx.x;          // 32*32*32 = 32768
  int lane = i & 31, cp = (i >> 5) & 31, nt = i >> 10;
  int k0 = 8 * cp + 2 * (lane >> 4);
  int n = nt * 16 + (lane & 15);
  v4f v;
  v.x = Rin * (kin[(size_t)(k0 + 0) * NN + n] - kin[(size_t)(k0 + 256) * NN + n]);
  v.y = Rin * (kin[(size_t)(k0 + 1) * NN + n] - kin[(size_t)(k0 + 257) * NN + n]);
  v.z = Rin * (kin[(size_t)(k0 + 4) * NN + n] - kin[(size_t)(k0 + 260) * NN + n]);
  v.w = Rin * (kin[(size_t)(k0 + 5) * NN + n] - kin[(size_t)(k0 + 261) * NN + n]);
  dst[i] = v;
}

__global__ __launch_bounds__(256) void pack_kout(const float* __restrict__ kout,
                                                 v4f* __restrict__ dst, float Rout) {
  int i = blockIdx.x * 256 + threadIdx.x;          // 8*64*32 = 16384
  int lane = i & 31, cp = (i >> 5) & 63, nt = i >> 11;
  int k0 = 8 * cp + 2 * (lane >> 4);
  int n = nt * 16 + (lane & 15);
  v4f v;
  v.x = Rout * kout[(size_t)(k0 + 0) * OUT_DIMS + n];
  v.y = Rout * kout[(size_t)(k0 + 1) * OUT_DIMS + n];
  v.z = Rout * kout[(size_t)(k0 + 4) * OUT_DIMS + n];
  v.w = Rout * kout[(size_t)(k0 + 5) * OUT_DIMS + n];
  dst[i] = v;
}

// ---- main SNN kernel: 1 WG = 16 batch rows x 512 neurons, all 33 steps ----
__global__ __launch_bounds__(256) void snn_step_kernel(
    const float* __restrict__ vm_in, const float* __restrict__ isyn_in,
    const float* __restrict__ rate_in, const float* __restrict__ spk_in,
    const float* __restrict__ x, const v4f* __restrict__ khp,
    const v4f* __restrict__ kinp, float* __restrict__ vm_out,
    float* __restrict__ isyn_out, float* __restrict__ rate_out,
    float* __restrict__ spk_out, float a_syn, float a_vm, float a_out) {
  __shared__ float lds[16 * NN];  // 32 KB: x staging, then spike buffer
  const int tid  = threadIdx.x;
  const int lane = tid & 31;
  const int wave = tid >> 5;      // 8 waves; wave owns neuron cols [64w, 64w+64)
  const int hi   = lane >> 4;     // A/B fragment half selector
  const int lrow = lane & 15;
  const int row0 = blockIdx.x * 16;

  // stage x tile [16][256] into LDS, coalesced
#pragma unroll
  for (int i = 0; i < 16; ++i) {
    int idx = i * 256 + tid;
    lds[idx] = x[(size_t)(row0 + (idx >> 8)) * IN_DIMS + (idx & 255)];
  }
  __syncthreads();

  // i_in = Rin * x @ (Win[:256]-Win[256:])  via WMMA over K=256
  v8f iin[4];
  {
    v8f acc[4];
#pragma unroll
    for (int t = 0; t < 4; ++t) acc[t] = v8zero();
    const v2f* arow = (const v2f*)(lds + lrow * IN_DIMS);
    for (int cp = 0; cp < 32; ++cp) {
      v2f a0 = arow[4 * cp + hi];      // chunk 2cp
      v2f a1 = arow[4 * cp + 2 + hi];  // chunk 2cp+1
#pragma unroll
      for (int t = 0; t < 4; ++t) {
        v4f b = kinp[((wave * 4 + t) * 32 + cp) * 32 + lane];  // b128
        acc[t] = wmma_f32(a0, __builtin_shufflevector(b, b, 0, 1), acc[t]);
        acc[t] = wmma_f32(a1, __builtin_shufflevector(b, b, 2, 3), acc[t]);
      }
    }
#pragma unroll
    for (int t = 0; t < 4; ++t) iin[t] = acc[t];
  }
  __syncthreads();  // done reading x; LDS becomes the spike buffer

  // load initial state into C-tile-layout registers
  v8f vmr[4], isr[4], rtr[4];
#pragma unroll
  for (int t = 0; t < 4; ++t) {
    const int nb = (wave * 4 + t) * 16 + lrow;
#pragma unroll
    for (int v = 0; v < 8; ++v) {
      size_t o = (size_t)(row0 + v + 8 * hi) * NN + nb;
      vmr[t][v] = vm_in[o];
      isr[t][v] = isyn_in[o];
      rtr[t][v] = rate_in[o];
    }
  }
  // initial spikes -> LDS [16][512], coalesced
#pragma unroll
  for (int i = 0; i < 32; ++i) {
    int idx = i * 256 + tid;
    lds[idx] = spk_in[(size_t)(row0 + (idx >> 9)) * NN + (idx & 511)];
  }

  for (int step = 0; step < NSTEPS; ++step) {
    __syncthreads();  // spike buffer for this step is complete
    v8f acc[4];
#pragma unroll
    for (int t = 0; t < 4; ++t) acc[t] = v8zero();
    const v2f* arow = (const v2f*)(lds + lrow * NN);
    for (int cp = 0; cp < 64; ++cp) {          // 2 K-chunks per iteration
      v2f a0 = arow[4 * cp + hi];              // ds_load_2addr_b64 pair
      v2f a1 = arow[4 * cp + 2 + hi];
#pragma unroll
      for (int t = 0; t < 4; ++t) {
        v4f b = khp[((wave * 4 + t) * 64 + cp) * 32 + lane];  // 1x b128, L2-res
        acc[t] = wmma_f32(a0, __builtin_shufflevector(b, b, 0, 1), acc[t]);
        acc[t] = wmma_f32(a1, __builtin_shufflevector(b, b, 2, 3), acc[t]);
      }
    }
    __syncthreads();  // everyone done reading old spikes

    // state update + write new spikes (acc already includes R scaling)
#pragma unroll
    for (int t = 0; t < 4; ++t) {
      const int nb = (wave * 4 + t) * 16 + lrow;
#pragma unroll
      for (int v = 0; v < 8; ++v) {
        float is  = isr[t][v] * a_syn + acc[t][v];
        isr[t][v] = is;
        float tgt = is + iin[t][v];
        float vmv = tgt + a_vm * (vmr[t][v] - tgt);   // lerp(v_m, tgt, a_vm)
        float s   = (vmv > 1.0f) ? 1.0f : 0.0f;      // heaviside(v_m - VTH)
        vmr[t][v] = vmv * (1.0f - s);                 // reset on spike
        float t2  = 2.0f * s;                         // (1/DT)*spike
        rtr[t][v] = t2 + a_out * (rtr[t][v] - t2);
        lds[(v + 8 * hi) * NN + nb] = s;
      }
    }
  }
  __syncthreads();

  // write final state
#pragma unroll
  for (int t = 0; t < 4; ++t) {
    const int nb = (wave * 4 + t) * 16 + lrow;
#pragma unroll
    for (int v = 0; v < 8; ++v) {
      size_t o = (size_t)(row0 + v + 8 * hi) * NN + nb;
      vm_out[o]   = vmr[t][v];
      isyn_out[o] = isr[t][v];
      rate_out[o] = rtr[t][v];
      spk_out[o]  = lds[(v + 8 * hi) * NN + nb];
    }
  }
}

// ---- out = Rout * rate @ kernel_out : [8192,512]x[512,128] ----
__global__ __launch_bounds__(256) void snn_out_kernel(
    const float* __restrict__ rate, const v4f* __restrict__ koutp,
    float* __restrict__ out) {
  __shared__ float lds[16 * NN];  // rate tile [16][512]
  const int tid  = threadIdx.x;
  const int lane = tid & 31;
  const int wave = tid >> 5;      // 8 waves -> 8 output N-tiles
  const int hi   = lane >> 4;
  const int lrow = lane & 15;
  const int row0 = blockIdx.x * 16;

#pragma unroll
  for (int i = 0; i < 32; ++i) {
    int idx = i * 256 + tid;
    lds[idx] = rate[(size_t)(row0 + (idx >> 9)) * NN + (idx & 511)];
  }
  __syncthreads();

  v8f acc = v8zero();
  const v2f* arow = (const v2f*)(lds + lrow * NN);
  for (int cp = 0; cp < 64; ++cp) {
    v2f a0 = arow[4 * cp + hi];
    v2f a1 = arow[4 * cp + 2 + hi];
    v4f b = koutp[(wave * 64 + cp) * 32 + lane];
    acc = wmma_f32(a0, __builtin_shufflevector(b, b, 0, 1), acc);
    acc = wmma_f32(a1, __builtin_shufflevector(b, b, 2, 3), acc);
  }
#pragma unroll
  for (int v = 0; v < 8; ++v) {
    out[(size_t)(row0 + v + 8 * hi) * OUT_DIMS + wave * 16 + lrow] = acc[v];
  }
}

extern "C" void kernel_launch(void* const* d_in, const int* in_sizes, int n_in,
                              void* d_out, int out_size, void* d_ws, size_t ws_size,
                              hipStream_t stream) {
  (void)in_sizes; (void)n_in; (void)out_size; (void)ws_size;
  const float* vm_in   = (const float*)d_in[0];
  const float* isyn_in = (const float*)d_in[1];
  const float* rate_in = (const float*)d_in[2];
  const float* spk_in  = (const float*)d_in[3];
  const float* x       = (const float*)d_in[4];
  const float* kin     = (const float*)d_in[5];
  const float* kh      = (const float*)d_in[6];
  const float* kout    = (const float*)d_in[7];

  float* out      = (float*)d_out;                       // [8192,128]
  float* vm_out   = out + (size_t)BATCH * OUT_DIMS;      // [8192,512]
  float* isyn_out = vm_out + (size_t)BATCH * NN;
  float* rate_out = isyn_out + (size_t)BATCH * NN;
  float* spk_out  = rate_out + (size_t)BATCH * NN;

  float* ws  = (float*)d_ws;
  v4f* khp   = (v4f*)(ws + KH_OFF);
  v4f* kinp  = (v4f*)(ws + KIN_OFF);
  v4f* koutp = (v4f*)(ws + KOUT_OFF);

  const float R_in  = (float)(0.1 * 1.0 * 10.0 * sqrt(2.0 / 256.0));
  const float R     = (float)(1.0 * 1.0 * (10.0 / 5.0) * sqrt(2.0 / 512.0));
  const float R_out = (float)(5.0 * sqrt(1.0 / 512.0));
  const float a_syn = (float)exp(-0.5 / 5.0);
  const float a_vm  = (float)exp(-0.5 / 10.0);
  const float a_out = (float)exp(-0.5 / 10.0);

  pack_kh  <<<256, 256, 0, stream>>>(kh,  khp,  R);
  pack_kin <<<128, 256, 0, stream>>>(kin, kinp, R_in);
  pack_kout<<< 64, 256, 0, stream>>>(kout, koutp, R_out);

  snn_step_kernel<<<BATCH / 16, 256, 0, stream>>>(
      vm_in, isyn_in, rate_in, spk_in, x, khp, kinp,
      vm_out, isyn_out, rate_out, spk_out, a_syn, a_vm, a_out);

  snn_out_kernel<<<BATCH / 16, 256, 0, stream>>>(rate_out, koutp, out);
}